// DiscriminativeLoss_24696061952722
// MI455X (gfx1250) — compile-verified
//
#include <hip/hip_runtime.h>
#include <math.h>

#define BATCH   16
#define DIMS    8
#define CLUST   32
#define NPIX    (512*512)
#define CHUNKS  64
#define THREADS 256

typedef float v2f __attribute__((ext_vector_type(2)));
typedef float v8f __attribute__((ext_vector_type(8)));

// Workspace layout (float offsets)
#define WS_SUMS    0        // B*C*D = 4096
#define WS_COUNTS  4096     // B*C   = 512
#define WS_CENTERS 4608     // B*C*D = 4096
#define WS_VARSUM  8704     // B
#define WS_DIST    8720     // B
#define WS_REG     8736     // B
#define WS_K       8752     // B
#define WS_TOTAL   8768

__global__ void k0_zero(float* ws) {
    int i = blockIdx.x * blockDim.x + threadIdx.x;
    if (i < WS_TOTAL) ws[i] = 0.0f;
}

// Pass 1: per-(batch,cluster) feature sums + counts.
// b128 streaming loads (4 pixels/lane/iter), per-wave replicated LDS
// accumulators, then global f32 atomics.
__global__ __launch_bounds__(THREADS) void k1_segsum(const float* __restrict__ data,
                                                     const int* __restrict__ labels,
                                                     float* __restrict__ ws) {
    __shared__ float acc[8][CLUST * (DIMS + 1)];   // 8 waves x 288 floats = 9216 B
    const int b   = blockIdx.y;
    const int wid = threadIdx.x >> 5;

    for (int i = threadIdx.x; i < 8 * CLUST * (DIMS + 1); i += THREADS)
        ((float*)acc)[i] = 0.0f;
    __syncthreads();

    const int pixPerBlock = NPIX / CHUNKS;         // 4096
    const int base = blockIdx.x * pixPerBlock;
    const float4* dptr4 = (const float4*)(data + (size_t)b * DIMS * NPIX);
    const int4*   lptr4 = (const int4*)(labels + (size_t)b * NPIX);

    for (int q = base / 4 + threadIdx.x; q < (base + pixPerBlock) / 4; q += THREADS) {
        int4 l = lptr4[q];
        const int c0 = l.x & (CLUST - 1), c1 = l.y & (CLUST - 1);
        const int c2 = l.z & (CLUST - 1), c3 = l.w & (CLUST - 1);
        float* aw = acc[wid];
#pragma unroll
        for (int d = 0; d < DIMS; ++d) {
            float4 x = dptr4[(size_t)d * (NPIX / 4) + q];   // 512B/wave per issue
            atomicAdd(&aw[c0 * (DIMS + 1) + d], x.x);
            atomicAdd(&aw[c1 * (DIMS + 1) + d], x.y);
            atomicAdd(&aw[c2 * (DIMS + 1) + d], x.z);
            atomicAdd(&aw[c3 * (DIMS + 1) + d], x.w);
        }
        atomicAdd(&aw[c0 * (DIMS + 1) + DIMS], 1.0f);
        atomicAdd(&aw[c1 * (DIMS + 1) + DIMS], 1.0f);
        atomicAdd(&aw[c2 * (DIMS + 1) + DIMS], 1.0f);
        atomicAdd(&aw[c3 * (DIMS + 1) + DIMS], 1.0f);
    }
    __syncthreads();

    for (int t = threadIdx.x; t < CLUST * (DIMS + 1); t += THREADS) {
        float s = 0.0f;
#pragma unroll
        for (int r = 0; r < 8; ++r) s += acc[r][t];
        int c = t / (DIMS + 1);
        int j = t % (DIMS + 1);
        if (j < DIMS) atomicAdd(&ws[WS_SUMS + (b * CLUST + c) * DIMS + j], s);
        else          atomicAdd(&ws[WS_COUNTS + b * CLUST + c], s);
    }
}

// Pass 2: centers, Gram via V_WMMA_F32_16X16X4_F32, dist hinge, reg, K.
// One wave per batch; 32x32 Gram = 2x2 tiles x 2 K-steps = 8 WMMAs.
__global__ __launch_bounds__(32) void k2_centers(float* __restrict__ ws) {
    __shared__ float cent[CLUST][DIMS];    // 1 KB
    __shared__ float gram[CLUST][CLUST];   // 4 KB
    __shared__ float presLds[CLUST];

    const int b = blockIdx.x;
    const int lane = threadIdx.x;

    // --- Phase A: centers + present + reg ---
    const int c = lane;
    float count = ws[WS_COUNTS + b * CLUST + c];
    bool present = count > 0.0f;
    float inv = 1.0f / fmaxf(count, 1.0f);
    float nsq = 0.0f;
#pragma unroll
    for (int d = 0; d < DIMS; ++d) {
        float v = ws[WS_SUMS + (b * CLUST + c) * DIMS + d] * inv;
        cent[c][d] = v;
        ws[WS_CENTERS + (b * CLUST + c) * DIMS + d] = v;
        nsq += v * v;
    }
    presLds[c] = present ? 1.0f : 0.0f;
    float regc = present ? fmaxf(sqrtf(nsq) - sqrtf((float)DIMS), 0.0f) : 0.0f;
    float K = (float)__popcll(__ballot(present));
#pragma unroll
    for (int off = 16; off > 0; off >>= 1) regc += __shfl_down(regc, off, 32);
    __syncthreads();

    // --- Phase B: Gram = c * c^T via WMMA (f32-exact) ---
    // A 16x4: row M = lane%16, K = 2*(lane/16)+{0,1}. B 4x16 mirrors with N = lane%16.
    const int m  = lane & 15;
    const int kk = (lane >> 4) << 1;       // 0 or 2
#pragma unroll
    for (int i0 = 0; i0 < CLUST; i0 += 16) {
#pragma unroll
        for (int j0 = 0; j0 < CLUST; j0 += 16) {
            v8f acc = {};
#pragma unroll
            for (int k0 = 0; k0 < DIMS; k0 += 4) {
                v2f av, bv;
                av.x = cent[i0 + m][k0 + kk];
                av.y = cent[i0 + m][k0 + kk + 1];
                bv.x = cent[j0 + m][k0 + kk];
                bv.y = cent[j0 + m][k0 + kk + 1];
                acc = __builtin_amdgcn_wmma_f32_16x16x4_f32(
                    /*neg_a=*/false, av, /*neg_b=*/false, bv,
                    /*c_mod=*/(short)0, acc, /*reuse_a=*/false, /*reuse_b=*/false);
            }
#pragma unroll
            for (int r = 0; r < 8; ++r) {
                int row = (lane < 16) ? r : (8 + r);
                gram[i0 + row][j0 + m] = acc[r];
            }
        }
    }
    __syncthreads();

    // --- Phase C: pairwise hinge from Gram ---
    float dsum = 0.0f;
    for (int t = lane; t < CLUST * CLUST; t += 32) {
        int i = t >> 5, j = t & 31;
        if (i != j && presLds[i] > 0.0f && presLds[j] > 0.0f) {
            float csq = gram[i][i] + gram[j][j] - 2.0f * gram[i][j];
            float cd = sqrtf(fmaxf(csq, 0.0f));
            float h = fmaxf(4.0f - cd, 0.0f);          // 2*DELTA_DIST = 4
            dsum += h * h;
        }
    }
#pragma unroll
    for (int off = 16; off > 0; off >>= 1) dsum += __shfl_down(dsum, off, 32);

    if (lane == 0) {
        ws[WS_K + b]    = K;
        ws[WS_REG + b]  = regc / K;
        ws[WS_DIST + b] = dsum * 0.5f / (K * fmaxf(K - 1.0f, 1.0f));
    }
}

// Pass 3: per-pixel variance hinge, summed per batch (cluster sums collapse).
// b128 streaming loads, 4 independent accumulation chains per thread.
__global__ __launch_bounds__(THREADS) void k3_var(const float* __restrict__ data,
                                                  const int* __restrict__ labels,
                                                  float* __restrict__ ws) {
    __shared__ float cent[CLUST][DIMS];
    __shared__ float total;
    const int b = blockIdx.y;
    if (threadIdx.x < CLUST * DIMS)
        ((float*)cent)[threadIdx.x] = ws[WS_CENTERS + b * CLUST * DIMS + threadIdx.x];
    if (threadIdx.x == 0) total = 0.0f;
    __syncthreads();

    const int pixPerBlock = NPIX / CHUNKS;
    const int base = blockIdx.x * pixPerBlock;
    const float4* dptr4 = (const float4*)(data + (size_t)b * DIMS * NPIX);
    const int4*   lptr4 = (const int4*)(labels + (size_t)b * NPIX);

    float hsum = 0.0f;
    for (int q = base / 4 + threadIdx.x; q < (base + pixPerBlock) / 4; q += THREADS) {
        int4 l = lptr4[q];
        const int c0 = l.x & (CLUST - 1), c1 = l.y & (CLUST - 1);
        const int c2 = l.z & (CLUST - 1), c3 = l.w & (CLUST - 1);
        float sq0 = 0.0f, sq1 = 0.0f, sq2 = 0.0f, sq3 = 0.0f;
#pragma unroll
        for (int d = 0; d < DIMS; ++d) {
            float4 x = dptr4[(size_t)d * (NPIX / 4) + q];
            float d0 = x.x - cent[c0][d]; sq0 += d0 * d0;
            float d1 = x.y - cent[c1][d]; sq1 += d1 * d1;
            float d2 = x.z - cent[c2][d]; sq2 += d2 * d2;
            float d3 = x.w - cent[c3][d]; sq3 += d3 * d3;
        }
        float t0 = fmaxf(((sq0 > 0.0f) ? sqrtf(sq0) : 0.0f) - 1.0f, 0.0f);  // DELTA_VAR = 1
        float t1 = fmaxf(((sq1 > 0.0f) ? sqrtf(sq1) : 0.0f) - 1.0f, 0.0f);
        float t2 = fmaxf(((sq2 > 0.0f) ? sqrtf(sq2) : 0.0f) - 1.0f, 0.0f);
        float t3 = fmaxf(((sq3 > 0.0f) ? sqrtf(sq3) : 0.0f) - 1.0f, 0.0f);
        hsum += t0 * t0 + t1 * t1 + t2 * t2 + t3 * t3;
    }
#pragma unroll
    for (int off = 16; off > 0; off >>= 1) hsum += __shfl_down(hsum, off, 32);
    if ((threadIdx.x & 31) == 0) atomicAdd(&total, hsum);
    __syncthreads();
    if (threadIdx.x == 0) atomicAdd(&ws[WS_VARSUM + b], total);
}

// Final combine + mean over batch.
__global__ __launch_bounds__(32) void k4_final(const float* __restrict__ ws,
                                               float* __restrict__ out) {
    const int lane = threadIdx.x;
    float v = 0.0f;
    if (lane < BATCH) {
        float K = ws[WS_K + lane];
        if (K > 1.0f) {
            float var_t = ws[WS_VARSUM + lane] / K;
            v = var_t + ws[WS_DIST + lane] + ws[WS_REG + lane];
        }
    }
#pragma unroll
    for (int off = 16; off > 0; off >>= 1) v += __shfl_down(v, off, 32);
    if (lane == 0) out[0] = v / (float)BATCH;
}

extern "C" void kernel_launch(void* const* d_in, const int* in_sizes, int n_in,
                              void* d_out, int out_size, void* d_ws, size_t ws_size,
                              hipStream_t stream) {
    const float* data   = (const float*)d_in[0];
    const int*   labels = (const int*)d_in[1];
    float* ws  = (float*)d_ws;
    float* out = (float*)d_out;

    k0_zero<<<(WS_TOTAL + 255) / 256, 256, 0, stream>>>(ws);
    dim3 grid(CHUNKS, BATCH);
    k1_segsum<<<grid, THREADS, 0, stream>>>(data, labels, ws);
    k2_centers<<<BATCH, 32, 0, stream>>>(ws);
    k3_var<<<grid, THREADS, 0, stream>>>(data, labels, ws);
    k4_final<<<1, 32, 0, stream>>>(ws, out);
}